// optimNet_25366076850571
// MI455X (gfx1250) — compile-verified
//
#include <hip/hip_runtime.h>
#include <hip/hip_bf16.h>

typedef float v2f __attribute__((ext_vector_type(2)));
typedef float v8f __attribute__((ext_vector_type(8)));
typedef unsigned int u32x4 __attribute__((ext_vector_type(4)));
typedef int i32x4 __attribute__((ext_vector_type(4)));
typedef int i32x8 __attribute__((ext_vector_type(8)));

// ---------------------------------------------------------------------------
// LDS A-panel geometry: 16 rows x KC f32, row stride KC+4 so the wave32
// ds_read_b64 fragment pattern hits all 64 banks conflict-free. The TDM pad
// feature reproduces this stride in hardware: pad 4 DWORDs every 256 DWORDs.
// ---------------------------------------------------------------------------
static constexpr int KC = 256;
static constexpr int LDS_STRIDE = KC + 4;

// Issue one TDM 2-D tile load: 16 x KC f32 tile of A[m0.., kc0..] -> LDS.
__device__ __forceinline__ void tdm_load_tile(const float* gtile, unsigned lds_off,
                                              unsigned rem_cols, unsigned rem_rows,
                                              unsigned stride_elems)
{
    unsigned long long ga = (unsigned long long)(uintptr_t)gtile;
    u32x4 g0;
    g0.x = 1u;                                      // count=1 (valid), user mode
    g0.y = lds_off;                                 // lds_addr (bytes)
    g0.z = (unsigned)(ga & 0xffffffffu);            // global_addr[31:0]
    g0.w = (unsigned)((ga >> 32) & 0x01ffffffu)     // global_addr[56:32]
         | (2u << 30);                              // type = 2 ("image")
    i32x8 g1;
    g1[0] = (int)((2u << 16)                        // data_size = 4 bytes
                | (1u << 20)                        // pad_enable
                | (7u << 22)                        // pad_interval: 256 DWORDs
                | (3u << 25));                      // pad_amount:   4 DWORDs
    g1[1] = (int)((rem_cols & 0xffffu) << 16);      // abar=0 | tensor_dim0[15:0]
    g1[2] = (int)((rem_cols >> 16) | ((rem_rows & 0xffffu) << 16)); // d0 hi | d1 lo
    g1[3] = (int)((rem_rows >> 16) | ((unsigned)KC << 16));         // d1 hi | tile_dim0
    g1[4] = 16;                                     // tile_dim1=16 | tile_dim2=0
    g1[5] = (int)stride_elems;                      // tensor_dim0_stride[31:0]
    g1[6] = 0;                                      // stride hi | dim1_stride lo
    g1[7] = 0;
    i32x4 z4 = {0, 0, 0, 0};
    i32x8 z8 = {0, 0, 0, 0, 0, 0, 0, 0};
    __builtin_amdgcn_tensor_load_to_lds(g0, g1, z4, z4, z8, 0);
}

// ---------------------------------------------------------------------------
// WMMA fp32 GEMM:  C[M x N] = op(A)[M x K] @ B[K x N]   (op = optional relu)
// Block = 256 threads (8 waves) owning 16 rows; waves split N into NSUB*16
// chunks. A staged by the Tensor Data Mover, double-buffered so the DMA of
// chunk i+1 overlaps the WMMA work on chunk i.
// ---------------------------------------------------------------------------
template <int K, int N, int NSUB, bool RELU>
__global__ __launch_bounds__(256) void gemm16_wmma(
    const float* __restrict__ A, const float* __restrict__ B,
    float* __restrict__ C, int M)
{
    __shared__ float As[2][16 * LDS_STRIDE];
    constexpr int NCHUNK = K / KC;

    const int lane = threadIdx.x & 31;
    const int wave = threadIdx.x >> 5;
    const int m0   = blockIdx.x << 4;
    const int col  = lane & 15;          // A-row / B-col / C-col within tile
    const int kp   = (lane >> 4) << 1;   // 0 or 2 (K sub-offset per ISA layout)
    const int n0   = wave * (NSUB * 16); // this wave's N origin

    const float*  Abase = A + (size_t)m0 * K;
    const unsigned rrows = (unsigned)(M - m0);

    if (wave == 0)
        tdm_load_tile(Abase, (unsigned)(uintptr_t)&As[0][0], (unsigned)K, rrows, (unsigned)K);

    v8f acc[NSUB] = {};

    for (int ci = 0; ci < NCHUNK; ++ci) {
        if (wave == 0)
            __builtin_amdgcn_s_wait_tensorcnt(0);   // chunk ci landed in LDS
        __syncthreads();                            // release compute waves
        if (wave == 0 && ci + 1 < NCHUNK)           // prefetch chunk ci+1
            tdm_load_tile(Abase + (ci + 1) * KC,
                          (unsigned)(uintptr_t)&As[(ci + 1) & 1][0],
                          (unsigned)(K - (ci + 1) * KC), rrows, (unsigned)K);

        const float* Ab = As[ci & 1];
        #pragma unroll 4
        for (int k4 = 0; k4 < KC; k4 += 4) {
            v2f a = *reinterpret_cast<const v2f*>(&Ab[col * LDS_STRIDE + k4 + kp]);
            if (RELU) { a.x = fmaxf(a.x, 0.f); a.y = fmaxf(a.y, 0.f); }
            const float* Bp = B + (size_t)(ci * KC + k4 + kp) * N + n0 + col;
            #pragma unroll
            for (int s = 0; s < NSUB; ++s) {
                v2f b;
                b.x = Bp[s * 16];        // row k+kp
                b.y = Bp[s * 16 + N];    // row k+kp+1
                acc[s] = __builtin_amdgcn_wmma_f32_16x16x4_f32(
                    false, a, false, b, (short)0, acc[s], false, false);
            }
        }
        __syncthreads();                            // buf reusable next issue
    }

    // ---- store C (16x16 f32 C/D layout: VGPR r -> rows r / r+8) ----
    const int gr_base = m0 + ((lane >> 4) << 3);
    #pragma unroll
    for (int s = 0; s < NSUB; ++s) {
        int gc = n0 + s * 16 + col;
        #pragma unroll
        for (int r = 0; r < 8; ++r) {
            int gr = gr_base + r;
            if (gr < M) C[(size_t)gr * N + gc] = acc[s][r];
        }
    }
}

// ---------------------------------------------------------------------------
// Degree / normalization helpers
// ---------------------------------------------------------------------------
__global__ void fill_ones(float* __restrict__ p, int n) {
    int i = blockIdx.x * blockDim.x + threadIdx.x;
    if (i < n) p[i] = 1.0f;                       // self-loop contribution
}

__global__ void scatter_deg(float* __restrict__ deg, const int* __restrict__ cols,
                            const float* __restrict__ w, int E) {
    int e = blockIdx.x * blockDim.x + threadIdx.x;
    if (e < E) atomicAdd(&deg[cols[e]], w[e]);
}

__global__ void finalize_dinv(float* __restrict__ deg, int n) {
    int i = blockIdx.x * blockDim.x + threadIdx.x;
    if (i < n) {
        float d = deg[i];
        deg[i] = (d > 0.f) ? rsqrtf(d) : 0.f;     // in-place deg -> dinv
    }
}

// ---------------------------------------------------------------------------
// out[i,:] = bias + xw[i,:] * dinv[i]^2   (self-loop term + bias, full init)
// ---------------------------------------------------------------------------
template <int F>
__global__ void init_out(float* __restrict__ out, const float* __restrict__ xw,
                         const float* __restrict__ dinv,
                         const float* __restrict__ bias, int Nn)
{
    int idx = blockIdx.x * blockDim.x + threadIdx.x;   // float4 units
    int total = Nn * (F / 4);
    if (idx >= total) return;
    int i  = idx / (F / 4);
    int j4 = idx - i * (F / 4);
    float d  = dinv[i];
    float dn = d * d;
    float4 v = *reinterpret_cast<const float4*>(xw + (size_t)idx * 4);
    float4 b = *reinterpret_cast<const float4*>(bias + j4 * 4);
    float4 o;
    o.x = b.x + v.x * dn; o.y = b.y + v.y * dn;
    o.z = b.z + v.z * dn; o.w = b.w + v.w * dn;
    *reinterpret_cast<float4*>(out + (size_t)idx * 4) = o;
}

// ---------------------------------------------------------------------------
// Edge aggregation:  out[col,:] += xw[row,:] * dinv[row]*w[e]*dinv[col]
// F/4 threads per edge, float4 gathers, f32 global atomics for scatter.
// ---------------------------------------------------------------------------
template <int F>
__global__ void edge_agg(const float* __restrict__ xw, const int* __restrict__ rows,
                         const int* __restrict__ cols, const float* __restrict__ w,
                         const float* __restrict__ dinv, float* __restrict__ out,
                         int E)
{
    constexpr int TPE = F / 4;
    int gid = blockIdx.x * blockDim.x + threadIdx.x;
    int e   = gid / TPE;
    if (e >= E) return;
    int t = gid - e * TPE;
    int r = rows[e];
    int c = cols[e];
    float norm = dinv[r] * w[e] * dinv[c];
    float4 v = *reinterpret_cast<const float4*>(xw + (size_t)r * F + t * 4);
    float* o = out + (size_t)c * F + t * 4;
    atomicAdd(o + 0, v.x * norm);
    atomicAdd(o + 1, v.y * norm);
    atomicAdd(o + 2, v.z * norm);
    atomicAdd(o + 3, v.w * norm);
}

// ---------------------------------------------------------------------------
// Similarity: s_src[i] = relu(h[i,:]) . Wm[0:512], s_dst[i] = . Wm[512:1024]
// One wave32 per node; shfl_xor tree reduction.
// ---------------------------------------------------------------------------
__global__ __launch_bounds__(256) void sim_scores(
    const float* __restrict__ h, const float* __restrict__ Wm,
    float* __restrict__ ssrc, float* __restrict__ sdst, int Nn)
{
    int lane = threadIdx.x & 31;
    int node = blockIdx.x * 8 + (threadIdx.x >> 5);
    if (node >= Nn) return;
    const float* hp = h + (size_t)node * 512;
    float a = 0.f, b = 0.f;
    #pragma unroll
    for (int it = 0; it < 4; ++it) {
        int j = lane * 4 + it * 128;
        float4 v  = *reinterpret_cast<const float4*>(hp + j);
        v.x = fmaxf(v.x, 0.f); v.y = fmaxf(v.y, 0.f);
        v.z = fmaxf(v.z, 0.f); v.w = fmaxf(v.w, 0.f);
        float4 ws = *reinterpret_cast<const float4*>(Wm + j);
        float4 wd = *reinterpret_cast<const float4*>(Wm + 512 + j);
        a += v.x * ws.x + v.y * ws.y + v.z * ws.z + v.w * ws.w;
        b += v.x * wd.x + v.y * wd.y + v.z * wd.z + v.w * wd.w;
    }
    #pragma unroll
    for (int off = 16; off > 0; off >>= 1) {
        a += __shfl_xor(a, off, 32);
        b += __shfl_xor(b, off, 32);
    }
    if (lane == 0) { ssrc[node] = a; sdst[node] = b; }
}

__global__ void e2_scores(const float* __restrict__ ssrc, const float* __restrict__ sdst,
                          const int* __restrict__ rows, const int* __restrict__ cols,
                          const float* __restrict__ bm, float* __restrict__ e2, int E)
{
    int e = blockIdx.x * blockDim.x + threadIdx.x;
    if (e < E) e2[e] = fmaxf(ssrc[rows[e]] + sdst[cols[e]] + bm[0], 0.f);
}

// ---------------------------------------------------------------------------
extern "C" void kernel_launch(void* const* d_in, const int* in_sizes, int n_in,
                              void* d_out, int out_size, void* d_ws, size_t ws_size,
                              hipStream_t stream)
{
    const float* node_attr = (const float*)d_in[0];
    const float* edge_attr = (const float*)d_in[1];
    const int*   ei        = (const int*)d_in[2];
    const float* W1 = (const float*)d_in[5];
    const float* b1 = (const float*)d_in[6];
    const float* W2 = (const float*)d_in[7];
    const float* b2 = (const float*)d_in[8];
    const float* Wm = (const float*)d_in[9];
    const float* bm = (const float*)d_in[10];

    const int F1 = in_sizes[6];            // 512
    const int F2 = in_sizes[8];            // 128
    const int K1 = in_sizes[5] / F1;       // 1024
    const int Nn = in_sizes[0] / K1;       // 100000
    const int E  = in_sizes[1];            // 1600000
    (void)F2; (void)n_in; (void)out_size; (void)ws_size;

    const int* rows = ei;
    const int* cols = ei + E;

    // ---- workspace carve (xw2 aliases xw1: xw1 dead after layer-1 agg) ----
    char* base = (char*)d_ws;
    size_t off = 0;
    auto carve = [&](size_t bytes) -> char* {
        char* p = base + off;
        off = (off + bytes + 255) & ~(size_t)255;
        return p;
    };
    float* xw1   = (float*)carve((size_t)Nn * F1 * sizeof(float));
    float* h     = (float*)carve((size_t)Nn * F1 * sizeof(float));
    float* dinv1 = (float*)carve((size_t)Nn * sizeof(float));
    float* dinv2 = (float*)carve((size_t)Nn * sizeof(float));
    float* ssrc  = (float*)carve((size_t)Nn * sizeof(float));
    float* sdst  = (float*)carve((size_t)Nn * sizeof(float));
    float* e2    = (float*)carve((size_t)E  * sizeof(float));
    float* xw2   = xw1;                     // alias
    float* out   = (float*)d_out;

    const int T = 256;
    auto cdiv = [](long long a, long long b) { return (int)((a + b - 1) / b); };

    // ---- layer 1 normalization ----
    fill_ones    <<<cdiv(Nn, T), T, 0, stream>>>(dinv1, Nn);
    scatter_deg  <<<cdiv(E, T),  T, 0, stream>>>(dinv1, cols, edge_attr, E);
    finalize_dinv<<<cdiv(Nn, T), T, 0, stream>>>(dinv1, Nn);

    // ---- GEMM1: xw1 = node_attr @ W1 ----
    gemm16_wmma<1024, 512, 4, false><<<cdiv(Nn, 16), T, 0, stream>>>(node_attr, W1, xw1, Nn);

    // ---- layer 1 aggregate: h = b1 + self-loop + edge scatter ----
    init_out<512><<<cdiv((long long)Nn * 128, T), T, 0, stream>>>(h, xw1, dinv1, b1, Nn);
    edge_agg<512><<<cdiv((long long)E * 128, T), T, 0, stream>>>(xw1, rows, cols, edge_attr, dinv1, h, E);

    // ---- similarity -> per-edge weights e2 ----
    sim_scores<<<cdiv(Nn, 8), T, 0, stream>>>(h, Wm, ssrc, sdst, Nn);
    e2_scores <<<cdiv(E, T),  T, 0, stream>>>(ssrc, sdst, rows, cols, bm, e2, E);

    // ---- layer 2 normalization ----
    fill_ones    <<<cdiv(Nn, T), T, 0, stream>>>(dinv2, Nn);
    scatter_deg  <<<cdiv(E, T),  T, 0, stream>>>(dinv2, cols, e2, E);
    finalize_dinv<<<cdiv(Nn, T), T, 0, stream>>>(dinv2, Nn);

    // ---- GEMM2: xw2 = relu(h) @ W2 (relu fused onto LDS A-fragments) ----
    gemm16_wmma<512, 128, 1, true><<<cdiv(Nn, 16), T, 0, stream>>>(h, W2, xw2, Nn);

    // ---- layer 2 aggregate straight into d_out ----
    init_out<128><<<cdiv((long long)Nn * 32, T), T, 0, stream>>>(out, xw2, dinv2, b2, Nn);
    edge_agg<128><<<cdiv((long long)E * 32, T), T, 0, stream>>>(xw2, rows, cols, e2, dinv2, out, E);
}